// Decoder_70729521431120
// MI455X (gfx1250) — compile-verified
//
#include <hip/hip_runtime.h>
#include <math.h>
#include <float.h>

#ifndef __has_builtin
#define __has_builtin(x) 0
#endif

#if __has_builtin(__builtin_amdgcn_wmma_f32_16x16x4_f32)
#define HAVE_WMMA_F32X4 1
#else
#define HAVE_WMMA_F32X4 0
#endif

#if __has_builtin(__builtin_amdgcn_global_load_async_to_lds_b128)
#define HAVE_ASYNC_LDS 1
#else
#define HAVE_ASYNC_LDS 0
#endif

#define B_SZ 4
#define CCH 9
#define ZD 64
#define YD 128
#define XD 128
#define NVOX (ZD * YD * XD)      /* 1048576 voxels per batch */
#define CHUNK 4096               /* floats per collect block  */
#define NCHUNK (NVOX / CHUNK)    /* 256 chunks per batch      */
#define CAP 2048                 /* candidate cap per batch   */
#define MC 1024                  /* M_CAND                    */
#define CLS_THR_C 0.5f
#define IOU_THR_C 0.5f

typedef float v2f __attribute__((ext_vector_type(2)));
typedef float v8f __attribute__((ext_vector_type(8)));

#if HAVE_ASYNC_LDS
// ROCm clang-22 declares the async-to-LDS builtin as
//   void (v4i addrspace(1)*, v4i addrspace(3)*, imm int, imm int)
// where v4i = int __attribute__((vector_size(16))).
typedef int v4i __attribute__((vector_size(16)));
typedef __attribute__((address_space(1))) v4i* gv4i_p;
typedef __attribute__((address_space(3))) v4i* lv4i_p;
#endif

__constant__ float c_mean[8] = {-0.25f, 0.01f, 0.26f, 0.46f, 0.26f, 0.62f, 0.69f, 1.46f};
__constant__ float c_std[8]  = { 0.93f, 0.26f, 0.53f, 0.89f, 1.11f, 0.13f, 0.16f, 0.19f};

// ---------------------------------------------------------------------------
// Kernel 0: zero the per-batch candidate counters.
// ---------------------------------------------------------------------------
__global__ void k_init(int* counts) {
  if (threadIdx.x < B_SZ) counts[threadIdx.x] = 0;
}

// ---------------------------------------------------------------------------
// Kernel 1: stream the cls channel (channel 0) only -- 16.8 MB instead of
// 151 MB -- staging each 16 KB chunk through LDS with async-to-LDS copies
// (CDNA5 GLOBAL_LOAD_ASYNC_TO_LDS_B128 + s_wait_asynccnt) when available.
// Threshold test + atomic compaction of (score, voxel index).
// ---------------------------------------------------------------------------
__global__ __launch_bounds__(256) void k_collect(const float* __restrict__ pts,
                                                 int* __restrict__ counts,
                                                 float* __restrict__ cscore,
                                                 int* __restrict__ cidx) {
  __shared__ float tile[CHUNK];
  const int b = blockIdx.y;
  const int chunk = blockIdx.x;
  const int tid = threadIdx.x;
  const float* src = pts + (size_t)b * CCH * NVOX + (size_t)chunk * CHUNK;

  if (chunk + 1 < NCHUNK) {
    __builtin_prefetch(src + CHUNK, 0, 1);   // global_prefetch_b8
  }

#if HAVE_ASYNC_LDS
  // 4 passes of 256 lanes x 16B = 16 KB into LDS, asynchronously.
  for (int p = 0; p < CHUNK / (4 * 256); ++p) {
    const int u = p * 256 + tid;             // 16-byte unit
    __builtin_amdgcn_global_load_async_to_lds_b128(
        (gv4i_p)(src + u * 4),
        (lv4i_p)(&tile[u * 4]),
        0, 0);
  }
#if __has_builtin(__builtin_amdgcn_s_wait_asynccnt)
  __builtin_amdgcn_s_wait_asynccnt(0);
#else
  asm volatile("s_wait_asynccnt 0" ::: "memory");
#endif
  __syncthreads();
#else
  for (int u = tid; u < CHUNK; u += 256) tile[u] = src[u];
  __syncthreads();
#endif

  for (int k = 0; k < 16; ++k) {
    const float v = tile[tid * 16 + k];
    if (v > CLS_THR_C) {
      const int gi = chunk * CHUNK + tid * 16 + k;
      const int pos = atomicAdd(&counts[b], 1);
      if (pos < CAP) {
        cscore[b * CAP + pos] = v;
        cidx[b * CAP + pos] = gi;
      }
    }
  }
}

// ---------------------------------------------------------------------------
// Kernel 2: one 1024-thread block (32 full wave32s) per batch.
//   - bitonic sort (score desc, idx asc) of up to CAP candidates
//   - decode top-1024 (gather 8 reg channels per candidate)
//   - corner AABBs: px/pz via V_WMMA_F32_16X16X4_F32 (A = [c, .5*ct*dim,
//     .5*st*dim, 0] per candidate row, B = [1; sx; sz; 0] per corner column),
//     then 8-lane shfl_xor min/max reductions over corner columns
//   - dual NMS (xy + yz projections) over the valid prefix
//   - prefix-sum compaction, emit top n_out kept boxes + scores
// ---------------------------------------------------------------------------
__global__ __launch_bounds__(MC) void k_perbatch(const float* __restrict__ pts,
                                                 const int* __restrict__ counts,
                                                 const float* __restrict__ cscore,
                                                 const int* __restrict__ cidx,
                                                 float* __restrict__ out_boxes,
                                                 float* __restrict__ out_scores,
                                                 int n_out) {
  extern __shared__ char smem[];
  float* s_score = (float*)smem;              // CAP
  int*   s_idx   = (int*)(s_score + CAP);     // CAP
  float* s_box   = (float*)(s_idx + CAP);     // MC*9
  float* s_xmin  = s_box + MC * 9;            // MC each below
  float* s_xmax  = s_xmin + MC;
  float* s_ymin  = s_xmax + MC;
  float* s_ymax  = s_ymin + MC;
  float* s_zmin  = s_ymax + MC;
  float* s_zmax  = s_zmin + MC;
  int* s_supxy   = (int*)(s_zmax + MC);       // MC
  int* s_supyz   = s_supxy + MC;              // MC
  int* s_pf      = s_supyz + MC;              // MC

  const int b = blockIdx.x;
  const int j = threadIdx.x;
  const int cnt_raw = counts[b];
  const int count = cnt_raw < CAP ? cnt_raw : CAP;

  // Zero this batch's output slice (d_out is poisoned by the harness).
  for (int t = j; t < n_out * 9; t += blockDim.x) out_boxes[(size_t)b * n_out * 9 + t] = 0.f;
  for (int t = j; t < n_out; t += blockDim.x)     out_scores[(size_t)b * n_out + t] = 0.f;

  // Load candidates (padding: score=-FLT_MAX, idx=0 so decode stays in-bounds).
  for (int t = j; t < CAP; t += blockDim.x) {
    if (t < count) {
      s_score[t] = cscore[b * CAP + t];
      s_idx[t]   = cidx[b * CAP + t];
    } else {
      s_score[t] = -FLT_MAX;
      s_idx[t]   = 0;
    }
  }
  __syncthreads();

  // Bitonic sort over CAP: score descending, ties by index ascending
  // (matches jax.lax.top_k tie order). 1024 threads handle 2048 slots.
  for (int k = 2; k <= CAP; k <<= 1) {
    for (int jj = k >> 1; jj > 0; jj >>= 1) {
      for (int t = j; t < CAP; t += blockDim.x) {
        const int l = t ^ jj;
        if (l > t) {
          const float s0 = s_score[t], s1 = s_score[l];
          const int i0 = s_idx[t], i1 = s_idx[l];
          const bool beforeTL = (s0 > s1) || (s0 == s1 && i0 < i1); // t-elem first in desc order
          const bool beforeLT = (s1 > s0) || (s1 == s0 && i1 < i0);
          const bool doswap = ((t & k) == 0) ? beforeLT : beforeTL;
          if (doswap) {
            s_score[t] = s1; s_score[l] = s0;
            s_idx[t] = i1;   s_idx[l] = i0;
          }
        }
      }
      __syncthreads();
    }
  }

  // Decode candidate j: gather 8 reg channels, build box9 and y-extents.
  {
    const int vidx = s_idx[j];
    const int z = vidx >> 14;        // / (128*128)
    const int rem = vidx & 16383;
    const int y = rem >> 7;
    const int x = rem & 127;
    const float xs = -16.f + 0.25f * (float)x;
    const float ys = -16.f + 0.25f * (float)y;
    const float zs = -16.f + 0.5f  * (float)z;
    const float* base = pts + (size_t)b * CCH * NVOX + vidx;
    float r[8];
#pragma unroll
    for (int c = 0; c < 8; ++c)
      r[c] = base[(size_t)(c + 1) * NVOX] * c_std[c] + c_mean[c];
    const float theta = atan2f(r[1], r[0]);
    const float cx = xs + r[2], cy = ys + r[3], cz = zs + r[4];
    const float w = expf(r[5]), h = expf(r[6]), l = expf(r[7]);
    s_box[j * 9 + 0] = cx;   s_box[j * 9 + 1] = cy;   s_box[j * 9 + 2] = cz;
    s_box[j * 9 + 3] = w;    s_box[j * 9 + 4] = h;    s_box[j * 9 + 5] = l;
    s_box[j * 9 + 6] = 0.f;  s_box[j * 9 + 7] = theta; s_box[j * 9 + 8] = 0.f;
    s_ymin[j] = cy - 0.5f * h;
    s_ymax[j] = cy + 0.5f * h;
    s_supxy[j] = 0;
    s_supyz[j] = 0;
  }
  __syncthreads();

#if HAVE_WMMA_F32X4
  // Corner x/z extents via WMMA: each wave processes two 16-candidate tiles.
  // ISA f32 16x4 A layout: lanes 0-15 hold {K0,K1}, lanes 16-31 hold {K2,K3};
  // B (4x16) by the same half-stride pattern: VGPR0={K0,K2}, VGPR1={K1,K3}.
  {
    const int lane = j & 31;
    const int wv = j >> 5;
    const int col = lane & 15;
    const int half = lane >> 4;
#pragma unroll
    for (int tpass = 0; tpass < 2; ++tpass) {
      const int base = (wv + tpass * 32) * 16;
      const int cand = base + col;
      const float cx = s_box[cand * 9 + 0], cz = s_box[cand * 9 + 2];
      const float w  = s_box[cand * 9 + 3], l  = s_box[cand * 9 + 5];
      const float th = s_box[cand * 9 + 7];
      float st, ct;
      sincosf(th, &st, &ct);
      v2f a_px, a_pz, bm;
      a_px.x = half ? (0.5f * st * l) : cx;              // K2 : K0
      a_px.y = half ? 0.f             : (0.5f * ct * w); // K3 : K1
      a_pz.x = half ? (0.5f * ct * l) : cz;
      a_pz.y = half ? 0.f             : (-0.5f * st * w);
      const float sx = (col & 4) ? -1.f : 1.f;
      const float sz = (col & 1) ? -1.f : 1.f;
      const bool on = col < 8;                           // corner columns 0..7
      bm.x = half ? (on ? sz : 0.f) : (on ? 1.f : 0.f);  // K2 row : K0 row
      bm.y = half ? 0.f             : (on ? sx : 0.f);   // K3 row : K1 row
      v8f c0 = {};
      v8f dpx = __builtin_amdgcn_wmma_f32_16x16x4_f32(false, a_px, false, bm,
                                                      (short)0, c0, false, false);
      v8f dpz = __builtin_amdgcn_wmma_f32_16x16x4_f32(false, a_pz, false, bm,
                                                      (short)0, c0, false, false);
      // D layout: VGPR r: lanes0-15 -> row r, lanes16-31 -> row r+8.
      // Reduce min/max over the 8 corner columns (stay inside 8-lane groups
      // so the zeroed columns 8..15 never pollute the result).
#pragma unroll
      for (int r = 0; r < 8; ++r) {
        float vxm = dpx[r], vxM = dpx[r];
        float vzm = dpz[r], vzM = dpz[r];
#pragma unroll
        for (int m = 1; m <= 4; m <<= 1) {
          vxm = fminf(vxm, __shfl_xor(vxm, m, 32));
          vxM = fmaxf(vxM, __shfl_xor(vxM, m, 32));
          vzm = fminf(vzm, __shfl_xor(vzm, m, 32));
          vzM = fmaxf(vzM, __shfl_xor(vzM, m, 32));
        }
        if (col == 0) {  // lanes 0 (rows r) and 16 (rows r+8)
          const int cc = base + half * 8 + r;
          s_xmin[cc] = vxm; s_xmax[cc] = vxM;
          s_zmin[cc] = vzm; s_zmax[cc] = vzM;
        }
      }
    }
  }
#else
  // Scalar fallback: 8 rotated corners per candidate.
  {
    const float cx = s_box[j * 9 + 0], cz = s_box[j * 9 + 2];
    const float w = s_box[j * 9 + 3], l = s_box[j * 9 + 5], th = s_box[j * 9 + 7];
    float st, ct;
    sincosf(th, &st, &ct);
    float xmn = FLT_MAX, xmx = -FLT_MAX, zmn = FLT_MAX, zmx = -FLT_MAX;
#pragma unroll
    for (int c = 0; c < 8; ++c) {
      const float sx = (c & 4) ? -1.f : 1.f;
      const float sz = (c & 1) ? -1.f : 1.f;
      const float ox = 0.5f * sx * w, oz = 0.5f * sz * l;
      const float px = cx + ct * ox + st * oz;
      const float pz = cz - st * ox + ct * oz;
      xmn = fminf(xmn, px); xmx = fmaxf(xmx, px);
      zmn = fminf(zmn, pz); zmx = fmaxf(zmx, pz);
    }
    s_xmin[j] = xmn; s_xmax[j] = xmx; s_zmin[j] = zmn; s_zmax[j] = zmx;
  }
#endif
  __syncthreads();

  // Dual NMS (xy and yz projections) over the valid (sorted) prefix.
  const float axm = s_xmin[j], axM = s_xmax[j];
  const float aym = s_ymin[j], ayM = s_ymax[j];
  const float azm = s_zmin[j], azM = s_zmax[j];
  const float areaXY = fmaxf(axM - axm, 0.f) * fmaxf(ayM - aym, 0.f);
  const float areaYZ = fmaxf(ayM - aym, 0.f) * fmaxf(azM - azm, 0.f);

  for (int i = 0; i < MC; ++i) {
    if (s_score[i] <= CLS_THR_C) break;     // sorted desc -> uniform break
    const bool kxy = (s_supxy[i] == 0);     // keep decision for box i
    const bool kyz = (s_supyz[i] == 0);
    if (j > i && (kxy || kyz)) {
      const float bxm = s_xmin[i], bxM = s_xmax[i];
      const float bym = s_ymin[i], byM = s_ymax[i];
      const float bzm = s_zmin[i], bzM = s_zmax[i];
      if (kxy) {
        const float iw = fminf(axM, bxM) - fmaxf(axm, bxm);
        const float ih = fminf(ayM, byM) - fmaxf(aym, bym);
        const float inter = fmaxf(iw, 0.f) * fmaxf(ih, 0.f);
        const float areaB = fmaxf(bxM - bxm, 0.f) * fmaxf(byM - bym, 0.f);
        const float uni = areaXY + areaB - inter;
        if (inter / fmaxf(uni, 1e-9f) > IOU_THR_C) s_supxy[j] = 1;
      }
      if (kyz) {
        const float iw = fminf(ayM, byM) - fmaxf(aym, bym);
        const float ih = fminf(azM, bzM) - fmaxf(azm, bzm);
        const float inter = fmaxf(iw, 0.f) * fmaxf(ih, 0.f);
        const float areaB = fmaxf(byM - bym, 0.f) * fmaxf(bzM - bzm, 0.f);
        const float uni = areaYZ + areaB - inter;
        if (inter / fmaxf(uni, 1e-9f) > IOU_THR_C) s_supyz[j] = 1;
      }
    }
    __syncthreads();
  }
  __syncthreads();

  // keep = valid & (keep_xy | keep_yz); stable compaction by prefix sum.
  const bool validj = s_score[j] > CLS_THR_C;
  const bool keep = validj && ((s_supxy[j] == 0) || (s_supyz[j] == 0));
  s_pf[j] = keep ? 1 : 0;
  __syncthreads();
  for (int off = 1; off < MC; off <<= 1) {
    const int t = (j >= off) ? s_pf[j - off] : 0;
    __syncthreads();
    s_pf[j] += t;
    __syncthreads();
  }
  const int pos = s_pf[j] - (keep ? 1 : 0);
  if (keep && pos < n_out) {
#pragma unroll
    for (int kk = 0; kk < 9; ++kk)
      out_boxes[((size_t)b * n_out + pos) * 9 + kk] = s_box[j * 9 + kk];
    out_scores[(size_t)b * n_out + pos] = s_score[j];
  }
}

// ---------------------------------------------------------------------------
extern "C" void kernel_launch(void* const* d_in, const int* in_sizes, int n_in,
                              void* d_out, int out_size, void* d_ws, size_t ws_size,
                              hipStream_t stream) {
  (void)in_sizes; (void)n_in; (void)ws_size;
  const float* pts = (const float*)d_in[0];
  const int n_out = out_size / (B_SZ * 10);  // boxes(9)+score(1) per slot

  float* out_boxes = (float*)d_out;
  float* out_scores = (float*)d_out + (size_t)B_SZ * n_out * 9;

  // Workspace layout: [counts: 256B pad] [cscore: B*CAP f32] [cidx: B*CAP i32]
  int* counts = (int*)d_ws;
  float* cscore = (float*)((char*)d_ws + 256);
  int* cidx = (int*)((char*)d_ws + 256 + (size_t)B_SZ * CAP * sizeof(float));

  k_init<<<1, 32, 0, stream>>>(counts);

  dim3 g1(NCHUNK, B_SZ);
  k_collect<<<g1, 256, 0, stream>>>(pts, counts, cscore, cidx);

  const size_t lds_bytes = (size_t)CAP * 4 * 2      // score + idx
                         + (size_t)MC * 9 * 4       // box9
                         + (size_t)MC * 4 * 6       // x/y/z extents
                         + (size_t)MC * 4 * 3;      // sup_xy, sup_yz, prefix
  k_perbatch<<<B_SZ, MC, lds_bytes, stream>>>(pts, counts, cscore, cidx,
                                              out_boxes, out_scores, n_out);
}